// LuongAttention_30064771072311
// MI455X (gfx1250) — compile-verified
//
#include <hip/hip_runtime.h>
#include <hip/hip_bf16.h>

// Problem constants (from reference): B=64, S=8192, H=256, E=256
#define BB   64
#define SS   8192
#define HH   256
#define EE   256
#define NCH  16                       // S-chunks per batch row
#define ROWS_PER_BLOCK (SS / NCH)     // 512
#define NWAVES 8
#define ROWS_PER_WAVE (ROWS_PER_BLOCK / NWAVES) // 64
#define GRP  4                        // rows per wave iteration (held in regs)
#define NEG_BIG (-3.0e38f)

typedef float v2f __attribute__((ext_vector_type(2)));
typedef float v8f __attribute__((ext_vector_type(8)));

// ---------------------------------------------------------------------------
// Kernel 1: v[b,e] = sum_h h[b,h] * W_a[h,e]   (M=64, N=256, K=256 f32 GEMM)
// One wave per 16x16 D tile, K stepped by 4 via V_WMMA_F32_16X16X4_F32.
// ---------------------------------------------------------------------------
#if __has_builtin(__builtin_amdgcn_wmma_f32_16x16x4_f32)
#define HAS_WMMA_F32 1
__global__ __launch_bounds__(32)
void compute_v_wmma(const float* __restrict__ h, const float* __restrict__ W,
                    float* __restrict__ v) {
  const int m0 = blockIdx.x * 16;     // over B rows
  const int n0 = blockIdx.y * 16;     // over E cols
  const int lane = threadIdx.x;
  const int hi   = lane >> 4;         // lane half (0: K0/K1, 1: K2/K3)
  const int lp   = lane & 15;
  const int row  = m0 + lp;           // A-matrix row (ISA 16x4 f32 layout)
  const int col  = n0 + lp;           // B/D column

  v8f c = {};
  for (int k = 0; k < HH; k += 4) {
    v2f a, b;
    const int kk = k + hi * 2;
    a.x = h[row * HH + kk + 0];
    a.y = h[row * HH + kk + 1];
    b.x = W[(kk + 0) * EE + col];
    b.y = W[(kk + 1) * EE + col];
    c = __builtin_amdgcn_wmma_f32_16x16x4_f32(
        /*neg_a=*/false, a, /*neg_b=*/false, b,
        /*c_mod=*/(short)0, c, /*reuse_a=*/false, /*reuse_b=*/false);
  }
#pragma unroll
  for (int r = 0; r < 8; ++r)         // D: VGPR r -> row r (lanes 0-15), r+8 (16-31)
    v[(m0 + r + hi * 8) * EE + col] = c[r];
}
#else
#define HAS_WMMA_F32 0
__global__ __launch_bounds__(256)
void compute_v_scalar(const float* __restrict__ h, const float* __restrict__ W,
                      float* __restrict__ v) {
  int idx = blockIdx.x * 256 + threadIdx.x;   // b*E + e
  int b = idx / EE, e = idx % EE;
  float acc = 0.f;
  for (int k = 0; k < HH; ++k) acc += h[b * HH + k] * W[k * EE + e];
  v[idx] = acc;
}
#endif

// ---------------------------------------------------------------------------
// Kernel 2: single streaming pass over encoder_outputs.
// Block (b, chunk): 512 rows. Each wave: 64 rows in groups of 4, online
// softmax; lane owns e = lane*8..lane*8+7 (coalesced float4 pairs).
// Emits raw masked scores + per-chunk (m, l, ctx[256]) partials.
// ---------------------------------------------------------------------------
__global__ __launch_bounds__(256)
void attn_pass(const float* __restrict__ enc, const unsigned char* __restrict__ mask,
               const float* __restrict__ v, float* __restrict__ scores,
               float* __restrict__ part) {
  const int b    = blockIdx.x;
  const int ch   = blockIdx.y;
  const int tid  = threadIdx.x;
  const int w    = tid >> 5;
  const int lane = tid & 31;

  __shared__ float lds_ctx[NWAVES][EE];
  __shared__ float lds_m[NWAVES];
  __shared__ float lds_l[NWAVES];

  float vr[8];
#pragma unroll
  for (int k = 0; k < 8; ++k) vr[k] = v[b * EE + lane * 8 + k];

  float m = NEG_BIG, l = 0.f;
  float ctx[8];
#pragma unroll
  for (int k = 0; k < 8; ++k) ctx[k] = 0.f;

  const int sblock = ch * ROWS_PER_BLOCK + w * ROWS_PER_WAVE;
  const float4* encb = (const float4*)(enc + (size_t)b * SS * EE);

  for (int it = 0; it < ROWS_PER_WAVE / GRP; ++it) {
    const int sbase = sblock + it * GRP;

    float4 a0[GRP], a1[GRP];
    float  sc[GRP];
#pragma unroll
    for (int r = 0; r < GRP; ++r) {
      const float4* rowp = encb + (size_t)(sbase + r) * (EE / 4);
      a0[r] = rowp[lane * 2];
      a1[r] = rowp[lane * 2 + 1];
      float dp = a0[r].x * vr[0] + a0[r].y * vr[1] + a0[r].z * vr[2] + a0[r].w * vr[3]
               + a1[r].x * vr[4] + a1[r].y * vr[5] + a1[r].z * vr[6] + a1[r].w * vr[7];
#pragma unroll
      for (int off = 16; off > 0; off >>= 1) dp += __shfl_xor(dp, off);
      sc[r] = dp;                       // broadcast on all lanes
    }
    if (it + 1 < ROWS_PER_WAVE / GRP)   // global_prefetch_b8 hint for next group
      __builtin_prefetch(encb + (size_t)(sbase + GRP) * (EE / 4) + lane * 2, 0, 1);

    const unsigned m4 = *(const unsigned*)(mask + (size_t)b * SS + sbase);
#pragma unroll
    for (int r = 0; r < GRP; ++r)
      if (((m4 >> (8 * r)) & 0xffu) == 0u) sc[r] = NEG_BIG;

    if (lane < GRP) {                   // raw masked scores -> scratch
      float out = sc[0];
#pragma unroll
      for (int r = 1; r < GRP; ++r) if (lane == r) out = sc[r];
      scores[(size_t)b * SS + sbase + lane] = out;
    }

    float mloc = sc[0];
#pragma unroll
    for (int r = 1; r < GRP; ++r) mloc = fmaxf(mloc, sc[r]);
    const float mnew = fmaxf(m, mloc);
    const float corr = __expf(m - mnew);
    l *= corr;
#pragma unroll
    for (int k = 0; k < 8; ++k) ctx[k] *= corr;

    float p[GRP];
#pragma unroll
    for (int r = 0; r < GRP; ++r) {
      p[r] = (sc[r] <= NEG_BIG) ? 0.f : __expf(sc[r] - mnew);
      l += p[r];
    }
#pragma unroll
    for (int r = 0; r < GRP; ++r) {
      ctx[0] += p[r] * a0[r].x; ctx[1] += p[r] * a0[r].y;
      ctx[2] += p[r] * a0[r].z; ctx[3] += p[r] * a0[r].w;
      ctx[4] += p[r] * a1[r].x; ctx[5] += p[r] * a1[r].y;
      ctx[6] += p[r] * a1[r].z; ctx[7] += p[r] * a1[r].w;
    }
    m = mnew;
  }

#pragma unroll
  for (int k = 0; k < 8; ++k) lds_ctx[w][lane * 8 + k] = ctx[k];
  if (lane == 0) { lds_m[w] = m; lds_l[w] = l; }
  __syncthreads();

  float mb = lds_m[0];
#pragma unroll
  for (int ww = 1; ww < NWAVES; ++ww) mb = fmaxf(mb, lds_m[ww]);
  float lb = 0.f, cb = 0.f;
#pragma unroll
  for (int ww = 0; ww < NWAVES; ++ww) {
    const float cf = __expf(lds_m[ww] - mb);
    lb += lds_l[ww] * cf;
    cb += lds_ctx[ww][tid] * cf;
  }
  float* pr = part + ((size_t)b * NCH + ch) * (EE + 2);
  if (tid == 0) { pr[0] = mb; pr[1] = lb; }
  pr[2 + tid] = cb;
}

// ---------------------------------------------------------------------------
// Kernel 3: combine the 16 chunk-partials per b -> context (normalized) + (m,l)
// ---------------------------------------------------------------------------
__global__ __launch_bounds__(256)
void combine_k(const float* __restrict__ part, float* __restrict__ out_ctx,
               float* __restrict__ ml) {
  const int b = blockIdx.x;
  const int t = threadIdx.x;
  float mg = NEG_BIG;
  for (int c = 0; c < NCH; ++c)
    mg = fmaxf(mg, part[((size_t)b * NCH + c) * (EE + 2)]);
  float lg = 0.f, cg = 0.f;
  for (int c = 0; c < NCH; ++c) {
    const float* pr = part + ((size_t)b * NCH + c) * (EE + 2);
    const float cf = __expf(pr[0] - mg);
    lg += pr[1] * cf;
    cg += pr[2 + t] * cf;
  }
  out_ctx[(size_t)b * EE + t] = cg / lg;
  if (t == 0) { ml[b * 2] = mg; ml[b * 2 + 1] = lg; }
}

// ---------------------------------------------------------------------------
// Kernel 4: attention_weights[b,s] = exp(score - m_b) / l_b
// ---------------------------------------------------------------------------
__global__ __launch_bounds__(256)
void weights_k(const float* __restrict__ scores, const float* __restrict__ ml,
               float* __restrict__ out_w) {
  const size_t idx = (size_t)blockIdx.x * 256 + threadIdx.x;
  const int b = (int)(idx >> 13);   // / 8192
  const float mg = ml[b * 2];
  const float lg = ml[b * 2 + 1];
  out_w[idx] = __expf(scores[idx] - mg) / lg;
}

// ---------------------------------------------------------------------------
extern "C" void kernel_launch(void* const* d_in, const int* in_sizes, int n_in,
                              void* d_out, int out_size, void* d_ws, size_t ws_size,
                              hipStream_t stream) {
  (void)in_sizes; (void)n_in; (void)out_size; (void)ws_size;
  const float* h            = (const float*)d_in[0];          // [B,H]
  const float* enc          = (const float*)d_in[1];          // [B,S,E]
  const unsigned char* mask = (const unsigned char*)d_in[2];  // [B,S] bool
  const float* Wa           = (const float*)d_in[3];          // [H,E]

  float* out_ctx = (float*)d_out;                 // [B,E]
  float* out_w   = out_ctx + (size_t)BB * EE;     // [B,S]

  float* ws        = (float*)d_ws;
  float* v_ws      = ws;                                      // B*E
  float* scores_ws = v_ws + (size_t)BB * EE;                  // B*S
  float* part_ws   = scores_ws + (size_t)BB * SS;             // B*NCH*(E+2)
  float* ml_ws     = part_ws + (size_t)BB * NCH * (EE + 2);   // B*2

#if HAS_WMMA_F32
  compute_v_wmma<<<dim3(BB / 16, EE / 16), 32, 0, stream>>>(h, Wa, v_ws);
#else
  compute_v_scalar<<<dim3((BB * EE) / 256), 256, 0, stream>>>(h, Wa, v_ws);
#endif
  attn_pass<<<dim3(BB, NCH), 256, 0, stream>>>(enc, mask, v_ws, scores_ws, part_ws);
  combine_k<<<BB, 256, 0, stream>>>(part_ws, out_ctx, ml_ws);
  weights_k<<<(BB * SS) / 256, 256, 0, stream>>>(scores_ws, ml_ws, out_w);
}